// SageGCN_49752901156947
// MI455X (gfx1250) — compile-verified
//
#include <hip/hip_runtime.h>
#include <hip/hip_bf16.h>
#include <stdint.h>

#define SCAN_BLOCK 1024
#define WAVES_PER_BLOCK 4
#define MAX_ROWS 32            // reference: action in [0,31] -> seg_len in [1,32]
#define DFEAT 128
#define ROW_BYTES (DFEAT * 4)  // 512 B per feature row

// ---------------------------------------------------------------------------
// Kernel 1: per-block inclusive scan of (action[i]+1); writes block-local
// exclusive prefix into seg_start and the block total into block_sums.
// ---------------------------------------------------------------------------
__global__ void sage_scan1(const int* __restrict__ action,
                           int* __restrict__ seg_start,
                           int* __restrict__ block_sums, int N) {
    __shared__ int sh[SCAN_BLOCK];
    const int t = threadIdx.x;
    const int i = blockIdx.x * SCAN_BLOCK + t;
    const int x = (i < N) ? (action[i] + 1) : 0;
    sh[t] = x;
    __syncthreads();
    for (int off = 1; off < SCAN_BLOCK; off <<= 1) {
        int v = (t >= off) ? sh[t - off] : 0;
        __syncthreads();
        sh[t] += v;
        __syncthreads();
    }
    if (i < N) seg_start[i] = sh[t] - x;             // exclusive within block
    if (t == SCAN_BLOCK - 1) block_sums[blockIdx.x] = sh[t];
}

// ---------------------------------------------------------------------------
// Kernel 2: sequential exclusive scan of the ~98 block sums (deterministic).
// ---------------------------------------------------------------------------
__global__ void sage_scan2(int* __restrict__ block_sums, int nb) {
    if (blockIdx.x == 0 && threadIdx.x == 0) {
        int run = 0;
        for (int b = 0; b < nb; ++b) {
            int v = block_sums[b];
            block_sums[b] = run;
            run += v;
        }
    }
}

// ---------------------------------------------------------------------------
// Kernel 3: add block offsets -> global exclusive prefix (seg_start).
// ---------------------------------------------------------------------------
__global__ void sage_scan3(int* __restrict__ seg_start,
                           const int* __restrict__ block_sums, int N) {
    const int i = blockIdx.x * SCAN_BLOCK + threadIdx.x;
    if (i < N) seg_start[i] += block_sums[blockIdx.x];
}

// ---------------------------------------------------------------------------
// Main kernel: one wave32 per node. Async-stage all segment rows into LDS
// (<= 32 outstanding GLOBAL_LOAD_ASYNC_TO_LDS_B128 per wave), wait once on
// ASYNCcnt, reduce from LDS, fuse mean + src + 0.5 scale + relu, store f4.
// ---------------------------------------------------------------------------
__global__ void __launch_bounds__(WAVES_PER_BLOCK * 32)
sage_main(const int* __restrict__ action,
          const float* __restrict__ src,
          const float* __restrict__ nbr,
          const int* __restrict__ seg_start,
          float* __restrict__ out, int N) {
    __shared__ __align__(16) float smem[WAVES_PER_BLOCK][MAX_ROWS * DFEAT];

    const int lane  = threadIdx.x & 31;
    const int wslot = threadIdx.x >> 5;
    const int node  = blockIdx.x * WAVES_PER_BLOCK + wslot;  // wave-uniform
    if (node >= N) return;  // whole wave exits together; EXEC stays all-ones

    const int base = seg_start[node];
    int len = action[node] + 1;
    if (len > MAX_ROWS) len = MAX_ROWS;  // defensive clamp (LDS capacity)

    // Lane L owns bytes [L*16, L*16+16) of every 512B row.
    uint32_t lds = (uint32_t)(uintptr_t)(&smem[wslot][0]) + (uint32_t)(lane * 16);
    uint64_t ga  = (uint64_t)(uintptr_t)nbr
                 + (uint64_t)base * ROW_BYTES
                 + (uint64_t)(lane * 16);

    // Fire one async 512B row copy per segment row; <= 32 in flight (ASYNCcnt<=63).
    for (int r = 0; r < len; ++r) {
        asm volatile("global_load_async_to_lds_b128 %0, %1, off"
                     :: "v"(lds), "v"(ga)
                     : "memory");
        lds += ROW_BYTES;
        ga  += ROW_BYTES;
    }
    // Single wait: all rows for this node resident in LDS.
    asm volatile("s_wait_asynccnt 0x0" ::: "memory");

    // Reduce rows from LDS (each lane reads back exactly the 16B it staged).
    float ax = 0.f, ay = 0.f, az = 0.f, aw = 0.f;
    const float* rows = &smem[wslot][lane * 4];
    #pragma unroll 4
    for (int r = 0; r < len; ++r) {
        const float* p = rows + r * DFEAT;
        ax += p[0]; ay += p[1]; az += p[2]; aw += p[3];
    }

    const float inv = 1.0f / (float)len;
    const size_t o  = (size_t)node * DFEAT + (size_t)(lane * 4);
    const float4 s  = *(const float4*)(src + o);
    float4 h;
    h.x = fmaxf(0.0f, (s.x + ax * inv) * 0.5f);
    h.y = fmaxf(0.0f, (s.y + ay * inv) * 0.5f);
    h.z = fmaxf(0.0f, (s.z + az * inv) * 0.5f);
    h.w = fmaxf(0.0f, (s.w + aw * inv) * 0.5f);
    *(float4*)(out + o) = h;
}

// ---------------------------------------------------------------------------
// Host entry: inputs are [action(int), src(f32 N x 128), neighbors(f32 T x 128)]
// ---------------------------------------------------------------------------
extern "C" void kernel_launch(void* const* d_in, const int* in_sizes, int n_in,
                              void* d_out, int out_size, void* d_ws, size_t ws_size,
                              hipStream_t stream) {
    const int*   action = (const int*)d_in[0];
    const float* src    = (const float*)d_in[1];
    const float* nbr    = (const float*)d_in[2];
    float*       out    = (float*)d_out;
    const int N = in_sizes[0];

    const int nb = (N + SCAN_BLOCK - 1) / SCAN_BLOCK;
    int* seg_start  = (int*)d_ws;
    int* block_sums = seg_start + ((N + 255) & ~255);  // keep block_sums aligned

    sage_scan1<<<nb, SCAN_BLOCK, 0, stream>>>(action, seg_start, block_sums, N);
    sage_scan2<<<1, 1, 0, stream>>>(block_sums, nb);
    sage_scan3<<<nb, SCAN_BLOCK, 0, stream>>>(seg_start, block_sums, N);

    const int nblocks = (N + WAVES_PER_BLOCK - 1) / WAVES_PER_BLOCK;
    sage_main<<<nblocks, WAVES_PER_BLOCK * 32, 0, stream>>>(
        action, src, nbr, seg_start, out, N);
}